// TopKPool_36447092473998
// MI455X (gfx1250) — compile-verified
//
#include <hip/hip_runtime.h>
#include <stdint.h>

#define NEG_INF (-__builtin_inff())

namespace {
constexpr int Bn   = 32;      // batches
constexpr int Nn   = 16384;   // cells
constexpr int Fn   = 256;     // filters
constexpr int Kn   = 64;      // top-k
constexpr int TILE = 16;      // rows per async chunk (16 KB contiguous)
constexpr int NBUF = 4;       // staging ring depth (prefetch-ahead-3)
constexpr int NTHR = 256;

// Min-heap of Kn entries per thread, strided in LDS: heap[j*Fn + tid].
// Bank = tid % 64 -> conflict-free within a wave32. Returns the new root
// (current 64th-largest) so the caller never re-reads LDS for it.
__device__ __forceinline__ float heap_sift_insert(float* __restrict__ heap,
                                                  int tid, float v) {
  // Level 0 handled explicitly so we can capture the new root.
  float lv = heap[1 * Fn + tid];
  float rv = heap[2 * Fn + tid];
  int   c;
  float cv;
  if (rv < lv) { c = 2; cv = rv; } else { c = 1; cv = lv; }
  if (cv >= v) { heap[tid] = v; return v; }
  heap[tid] = cv;
  const float root = cv;
  int idx = c;
  #pragma unroll 1
  while (true) {
    int l = 2 * idx + 1;
    if (l >= Kn) break;
    int r = l + 1;
    lv = heap[l * Fn + tid];
    rv = (r < Kn) ? heap[r * Fn + tid] : __builtin_inff();
    if (rv < lv) { c = r; cv = rv; } else { c = l; cv = lv; }
    if (cv >= v) break;
    heap[idx * Fn + tid] = cv;
    idx = c;
  }
  heap[idx * Fn + tid] = v;
  return root;
}

// Issue one 16KB chunk (TILE rows x 256 floats) as 4 async b128 loads/thread.
// GVS mode: saddr(SGPR64 base) + vaddr(32-bit byte offset). Non-temporal:
// the 512MB x stream should not thrash the 192MB L2.
__device__ __forceinline__ void async_stage_chunk(uint32_t lds_base,
                                                  uint64_t gbase,
                                                  uint32_t gbyte_off,
                                                  int tid) {
  #pragma unroll
  for (int i = 0; i < 4; ++i) {
    uint32_t elem = (uint32_t)(tid + i * NTHR) * 16u;  // float4 slot
    uint32_t ldsa = lds_base + elem;
    uint32_t voff = gbyte_off + elem;
    asm volatile("global_load_async_to_lds_b128 %0, %1, %2 th:TH_LOAD_NT"
                 :: "v"(ldsa), "v"(voff), "s"(gbase)
                 : "memory");
  }
}
} // namespace

// Phase 1: per (batch=blockIdx.y, slice=blockIdx.x) exact top-64 per filter
// via streaming LDS heaps fed by a 4-deep async-to-LDS ring.
// fused==1 (gridDim.x==1): finalize averaged output directly.
extern "C" __global__ __launch_bounds__(NTHR) void topk_partial_kernel(
    const float* __restrict__ x, const uint8_t* __restrict__ mask,
    float* __restrict__ dst, int M, int C, int fused)
{
  extern __shared__ char smem[];
  float*   stage = (float*)smem;                                     // NBUF*TILE*Fn f32
  float*   heap  = (float*)(smem + NBUF * TILE * Fn * 4);            // Kn*Fn f32
  int*     red   = (int*)  (smem + NBUF * TILE * Fn * 4 + Kn * Fn * 4);
  uint8_t* lmask = (uint8_t*)(smem + NBUF * TILE * Fn * 4 + Kn * Fn * 4 + NTHR * 4);

  const int tid = threadIdx.x;
  const int s   = blockIdx.x;       // slice
  const int b   = blockIdx.y;       // batch
  const int S   = gridDim.x;
  const int n0  = s * M;

  // Stage this slice's mask bytes into LDS.
  const uint8_t* mrow = mask + (size_t)b * Nn + n0;
  for (int j = tid; j < M; j += NTHR) lmask[j] = mrow[j];

  // Init per-filter heap with -inf sentinels (own column only).
  #pragma unroll
  for (int j = 0; j < Kn; ++j) heap[j * Fn + tid] = NEG_INF;

  const uint64_t gbase =
      (uint64_t)(uintptr_t)(x + ((size_t)b * Nn + n0) * Fn);
  const uint32_t stage0 = (uint32_t)(uintptr_t)stage;
  constexpr uint32_t CHUNK_BYTES = (uint32_t)(TILE * Fn * 4);

  // Prologue: prefetch chunks 0..2.
  #pragma unroll
  for (int p = 0; p < NBUF - 1; ++p)
    if (p < C)
      async_stage_chunk(stage0 + (uint32_t)p * CHUNK_BYTES, gbase,
                        (uint32_t)p * CHUNK_BYTES, tid);

  __syncthreads();  // publish lmask; async data gated by waits below

  float heapMin = NEG_INF;
  for (int c = 0; c < C; ++c) {
    const int pf = c + (NBUF - 1);
    if (pf < C) {
      async_stage_chunk(stage0 + (uint32_t)(pf & (NBUF - 1)) * CHUNK_BYTES,
                        gbase, (uint32_t)pf * CHUNK_BYTES, tid);
    }
    // Async loads complete in order: leaving `ahead` chunk-groups (4 instr
    // each) outstanding guarantees chunk c has landed.
    const int ahead = C - 1 - c;
    if      (ahead >= 3) asm volatile("s_wait_asynccnt 12" ::: "memory");
    else if (ahead == 2) asm volatile("s_wait_asynccnt 8"  ::: "memory");
    else if (ahead == 1) asm volatile("s_wait_asynccnt 4"  ::: "memory");
    else                 asm volatile("s_wait_asynccnt 0"  ::: "memory");
    __syncthreads();  // all waves' chunk-c data visible

    const float* sb  = stage + (c & (NBUF - 1)) * TILE * Fn;
    const int    nb0 = c * TILE;
    // One b128 LDS read for the 16 mask bytes of this chunk.
    const uint4 mw = *(const uint4*)(lmask + nb0);
    const uint32_t mword[4] = {mw.x, mw.y, mw.z, mw.w};
    #pragma unroll
    for (int r = 0; r < TILE; ++r) {
      float v = sb[r * Fn + tid];
      const bool pad = ((mword[r >> 2] >> ((r & 3) * 8)) & 0xffu) != 0u;
      if (!pad && (v > heapMin)) heapMin = heap_sift_insert(heap, tid, v);
    }
    __syncthreads();  // done reading this buffer before it is re-staged
  }

  if (!fused) {
    // Emit partial top-64 list: layout [((b*S+s)*Kn + j)*Fn + f]
    float* p = dst + (size_t)(b * S + s) * Kn * Fn;
    #pragma unroll
    for (int j = 0; j < Kn; ++j) p[(size_t)j * Fn + tid] = heap[j * Fn + tid];
  } else {
    // Single-pass fallback: finalize directly.
    int cnt = 0;
    for (int j = tid; j < M; j += NTHR) cnt += (lmask[j] == 0) ? 1 : 0;
    red[tid] = cnt;
    __syncthreads();
    for (int off = NTHR / 2; off > 0; off >>= 1) {
      if (tid < off) red[tid] += red[tid + off];
      __syncthreads();
    }
    int nv = red[0];
    nv = nv < Kn ? nv : Kn;
    if (nv < 1) nv = 1;
    float sum = 0.f;
    #pragma unroll
    for (int j = 0; j < Kn; ++j) {
      float h = heap[j * Fn + tid];
      sum += (h > NEG_INF) ? h : 0.f;   // -inf sentinel -> 0 (matches reference)
    }
    dst[(size_t)b * Fn + tid] = sum / (float)nv;
  }
}

// Phase 2: merge S partial top-64 lists per (b,f), finalize average.
extern "C" __global__ __launch_bounds__(NTHR) void topk_merge_kernel(
    const float* __restrict__ part, const uint8_t* __restrict__ mask,
    float* __restrict__ out, int S)
{
  extern __shared__ char smem[];
  float* heap = (float*)smem;             // Kn*Fn f32
  int*   red  = (int*)(smem + Kn * Fn * 4);

  const int tid = threadIdx.x;
  const int b   = blockIdx.x;

  int cnt = 0;
  const uint8_t* mrow = mask + (size_t)b * Nn;
  for (int j = tid; j < Nn; j += NTHR) cnt += (mrow[j] == 0) ? 1 : 0;
  red[tid] = cnt;
  #pragma unroll
  for (int j = 0; j < Kn; ++j) heap[j * Fn + tid] = NEG_INF;
  __syncthreads();
  for (int off = NTHR / 2; off > 0; off >>= 1) {
    if (tid < off) red[tid] += red[tid + off];
    __syncthreads();
  }
  const int n_valid = red[0];

  const float* base = part + (size_t)b * S * Kn * Fn;
  float heapMin = NEG_INF;
  const int T = S * Kn;  // multiple of 4
  for (int t = 0; t < T; t += 4) {
    float v0 = base[(size_t)(t + 0) * Fn + tid];
    float v1 = base[(size_t)(t + 1) * Fn + tid];
    float v2 = base[(size_t)(t + 2) * Fn + tid];
    float v3 = base[(size_t)(t + 3) * Fn + tid];
    if (v0 > heapMin) heapMin = heap_sift_insert(heap, tid, v0);
    if (v1 > heapMin) heapMin = heap_sift_insert(heap, tid, v1);
    if (v2 > heapMin) heapMin = heap_sift_insert(heap, tid, v2);
    if (v3 > heapMin) heapMin = heap_sift_insert(heap, tid, v3);
  }

  float sum = 0.f;
  #pragma unroll
  for (int j = 0; j < Kn; ++j) {
    float h = heap[j * Fn + tid];
    sum += (h > NEG_INF) ? h : 0.f;
  }
  int nv = n_valid < Kn ? n_valid : Kn;
  if (nv < 1) nv = 1;
  out[(size_t)b * Fn + tid] = sum / (float)nv;
}

extern "C" void kernel_launch(void* const* d_in, const int* in_sizes, int n_in,
                              void* d_out, int out_size, void* d_ws, size_t ws_size,
                              hipStream_t stream) {
  (void)in_sizes; (void)n_in; (void)out_size;
  const float*   x    = (const float*)d_in[0];
  const uint8_t* mask = (const uint8_t*)d_in[1];  // jax bool -> 1 byte/elem
  float*         out  = (float*)d_out;

  const size_t perS = (size_t)Bn * Kn * Fn * sizeof(float);  // 2 MiB per slice-set
  int S = 16;
  while (S > 1 && (size_t)S * perS > ws_size) S >>= 1;
  const bool twophase = (d_ws != nullptr) && ((size_t)S * perS <= ws_size);

  const size_t shFixed =
      (size_t)NBUF * TILE * Fn * 4 + (size_t)Kn * Fn * 4 + NTHR * 4;

  if (twophase) {
    const int M = Nn / S;
    const int C = M / TILE;
    topk_partial_kernel<<<dim3(S, Bn), dim3(NTHR), shFixed + (size_t)M, stream>>>(
        x, mask, (float*)d_ws, M, C, 0);
    const size_t sh2 = (size_t)Kn * Fn * 4 + NTHR * 4;
    topk_merge_kernel<<<dim3(Bn), dim3(NTHR), sh2, stream>>>(
        (const float*)d_ws, mask, out, S);
  } else {
    const int M = Nn;
    const int C = M / TILE;
    topk_partial_kernel<<<dim3(1, Bn), dim3(NTHR), shFixed + (size_t)M, stream>>>(
        x, mask, out, M, C, 1);
  }
}